// encoder_bead_11218454577219
// MI455X (gfx1250) — compile-verified
//
#include <hip/hip_runtime.h>

#define NN 100000
#define EE 1600000
#define EPSF 1e-12f

typedef __attribute__((ext_vector_type(2))) float v2f;
typedef __attribute__((ext_vector_type(8))) float v8f;

// Hardware fp32 atomic add (global_atomic_add_f32), never a CAS loop.
__device__ __forceinline__ void atomAddF32(float* p, float v) {
  unsafeAtomicAdd(p, v);
}

// ---------------------------------------------------------------- utilities
__global__ void fill_zero_kernel(float* __restrict__ p, unsigned int n) {
  unsigned int i = blockIdx.x * blockDim.x + threadIdx.x;
  if (i < n) p[i] = 0.0f;
}

// outdeg[src] += w ; indeg[dst] += w  (EdgeWeightNorm 'both' degree sums)
__global__ void deg_kernel(const int* __restrict__ src, const int* __restrict__ dst,
                           const float* __restrict__ w, float* __restrict__ outdeg,
                           float* __restrict__ indeg, int ne) {
  int e = blockIdx.x * blockDim.x + threadIdx.x;
  if (e >= ne) return;
  float we = w[e];
  atomAddF32(&outdeg[src[e]], we);
  atomAddF32(&indeg[dst[e]], we);
}

// nw = w / sqrt(max(outdeg[src]*indeg[dst], eps)); deg[dst] += nw
__global__ void nw_kernel(const int* __restrict__ src, const int* __restrict__ dst,
                          const float* __restrict__ w, const float* __restrict__ outdeg,
                          const float* __restrict__ indeg, float* __restrict__ nw,
                          float* __restrict__ deg, int ne) {
  int e = blockIdx.x * blockDim.x + threadIdx.x;
  if (e >= ne) return;
  int s = src[e], d = dst[e];
  float pd = fmaxf(outdeg[s] * indeg[d], EPSF);
  float nwe = w[e] / sqrtf(pd);
  nw[e] = nwe;
  atomAddF32(&deg[d], nwe);
}

// agg[dst,:] += h[src,:] * nw   (one thread per (edge, 4-feature chunk))
template <int F>
__global__ void scatter_kernel(const int* __restrict__ src, const int* __restrict__ dst,
                               const float* __restrict__ nw, const float* __restrict__ h,
                               float* __restrict__ agg, int ne) {
  constexpr unsigned V = F / 4;
  unsigned int idx = blockIdx.x * blockDim.x + threadIdx.x;
  unsigned int e = idx / V;
  if (e >= (unsigned)ne) return;
  unsigned int f = (idx % V) * 4u;
  int s = src[e], d = dst[e];
  float nwe = nw[e];
  const float4 hv = *reinterpret_cast<const float4*>(h + (size_t)s * F + f);
  float* a = agg + (size_t)d * F + f;
  atomAddF32(a + 0, hv.x * nwe);
  atomAddF32(a + 1, hv.y * nwe);
  atomAddF32(a + 2, hv.z * nwe);
  atomAddF32(a + 3, hv.w * nwe);
}

// x[i,f] = x[i,f] / max(deg[i], eps) + b[f]
template <int F>
__global__ void normalize_kernel(float* __restrict__ x, const float* __restrict__ deg,
                                 const float* __restrict__ b, int n) {
  unsigned int idx = blockIdx.x * blockDim.x + threadIdx.x;
  if (idx >= (unsigned)(n * F)) return;
  unsigned int i = idx / F, f = idx % F;
  x[idx] = x[idx] / fmaxf(deg[i], EPSF) + b[f];
}

// ---------------------------------------------------------------- fp32 WMMA GEMM
// Y[n x KO] = X[n x K] @ W[K x KO], row-major, via V_WMMA_F32_16X16X4_F32.
// One wave computes 16 rows x KO cols. Requires n % 16 == 0, KO % 16 == 0, K % 4 == 0.
// A 16x4 layout  (ISA 7.12.2): lanes 0-15 hold M=lane  {K=kk,kk+1};
//                              lanes 16-31 hold M=lane-16 {K=kk+2,kk+3}.
// B 4x16 layout  (symmetric):  v0 = row kk(+2) striped over N=lane%16, v1 = row kk+1(+3).
// C/D 16x16:     VGPR r: lanes 0-15 -> (M=r, N=lane); lanes 16-31 -> (M=8+r, N=lane-16).
template <int K, int KO>
__global__ void gemm_wmma_kernel(const float* __restrict__ X, const float* __restrict__ W,
                                 float* __restrict__ Y, int n) {
  constexpr int NT = KO / 16;
  int wave = (int)((blockIdx.x * blockDim.x + threadIdx.x) >> 5);
  int lane = (int)(threadIdx.x & 31u);
  int m0 = wave * 16;
  if (m0 >= n) return;                    // wave-uniform: EXEC stays all-ones for WMMA
  int half  = lane >> 4;                  // 0 = lanes 0-15, 1 = lanes 16-31
  int lr    = lane & 15;
  int khalf = half * 2;

  const float* xr = X + (size_t)(m0 + lr) * K;
  v8f zero = {};
  v8f acc[NT];
#pragma unroll
  for (int t = 0; t < NT; ++t) acc[t] = zero;

  for (int kk = 0; kk < K; kk += 4) {
    v2f a;
    a.x = xr[kk + khalf];
    a.y = xr[kk + khalf + 1];
#pragma unroll
    for (int t = 0; t < NT; ++t) {
      const float* wc = W + (size_t)(kk + khalf) * KO + t * 16 + lr;
      v2f b;
      b.x = wc[0];
      b.y = wc[KO];
      acc[t] = __builtin_amdgcn_wmma_f32_16x16x4_f32(
          /*neg_a=*/false, a, /*neg_b=*/false, b,
          /*c_mod=*/(short)0, acc[t], /*reuse_a=*/false, /*reuse_b=*/false);
    }
  }

  int mrow = m0 + half * 8;
#pragma unroll
  for (int t = 0; t < NT; ++t) {
#pragma unroll
    for (int r = 0; r < 8; ++r) {
      Y[(size_t)(mrow + r) * KO + t * 16 + lr] = acc[t][r];
    }
  }
}

// ---------------------------------------------------------------- launcher
extern "C" void kernel_launch(void* const* d_in, const int* in_sizes, int n_in,
                              void* d_out, int out_size, void* d_ws, size_t ws_size,
                              hipStream_t stream) {
  const int N = NN, E = EE;
  const float* x       = (const float*)d_in[0];
  const int*   srcs[3] = {(const int*)d_in[1], (const int*)d_in[4], (const int*)d_in[7]};
  const int*   dsts[3] = {(const int*)d_in[2], (const int*)d_in[5], (const int*)d_in[8]};
  const float* wgt[3]  = {(const float*)d_in[3], (const float*)d_in[6], (const float*)d_in[9]};
  const float* Wm[3]   = {(const float*)d_in[10], (const float*)d_in[12], (const float*)d_in[14]};
  const float* bm[3]   = {(const float*)d_in[11], (const float*)d_in[13], (const float*)d_in[15]};
  float* out = (float*)d_out;

  // workspace: [outdeg N][indeg N][deg N][nw E][bufH 64N][bufA 64N]  (~59 MB)
  float* wsf    = (float*)d_ws;
  float* outdeg = wsf;
  float* indeg  = wsf + (size_t)N;
  float* deg    = wsf + (size_t)2 * N;
  float* nw     = wsf + (size_t)3 * N;
  float* bufH   = wsf + (size_t)3 * N + E;
  float* bufA   = bufH + (size_t)64 * N;

  const int T = 256;
  auto blk = [](long long n, int t) { return (unsigned)((n + t - 1) / t); };
  const unsigned gemmThreads = (unsigned)((N / 16) * 32);  // N % 16 == 0

  // ---------------- layer 1: 128 -> 64 (transform first) ----------------
  fill_zero_kernel<<<blk(3LL * N, T), T, 0, stream>>>(outdeg, (unsigned)(3 * N));
  deg_kernel<<<blk(E, T), T, 0, stream>>>(srcs[0], dsts[0], wgt[0], outdeg, indeg, E);
  nw_kernel<<<blk(E, T), T, 0, stream>>>(srcs[0], dsts[0], wgt[0], outdeg, indeg, nw, deg, E);
  gemm_wmma_kernel<128, 64><<<blk(gemmThreads, T), T, 0, stream>>>(x, Wm[0], bufH, N);
  fill_zero_kernel<<<blk(64LL * N, T), T, 0, stream>>>(bufA, (unsigned)(64 * N));
  scatter_kernel<64><<<blk((long long)E * 16, T), T, 0, stream>>>(srcs[0], dsts[0], nw, bufH, bufA, E);
  normalize_kernel<64><<<blk(64LL * N, T), T, 0, stream>>>(bufA, deg, bm[0], N);
  // bufA now holds h_after1 (N x 64)

  // ---------------- layer 2: 64 -> 32 (transform first) ----------------
  fill_zero_kernel<<<blk(3LL * N, T), T, 0, stream>>>(outdeg, (unsigned)(3 * N));
  deg_kernel<<<blk(E, T), T, 0, stream>>>(srcs[1], dsts[1], wgt[1], outdeg, indeg, E);
  nw_kernel<<<blk(E, T), T, 0, stream>>>(srcs[1], dsts[1], wgt[1], outdeg, indeg, nw, deg, E);
  gemm_wmma_kernel<64, 32><<<blk(gemmThreads, T), T, 0, stream>>>(bufA, Wm[1], bufH, N);
  float* h2   = bufH;                         // N x 32
  float* agg2 = bufH + (size_t)32 * N;        // N x 32, disjoint from h2
  fill_zero_kernel<<<blk(32LL * N, T), T, 0, stream>>>(agg2, (unsigned)(32 * N));
  scatter_kernel<32><<<blk((long long)E * 8, T), T, 0, stream>>>(srcs[1], dsts[1], nw, h2, agg2, E);
  normalize_kernel<32><<<blk(32LL * N, T), T, 0, stream>>>(agg2, deg, bm[1], N);
  // agg2 now holds h_after2 (N x 32)

  // ---------------- layer 3: 32 -> 32 (aggregate first) ----------------
  fill_zero_kernel<<<blk(3LL * N, T), T, 0, stream>>>(outdeg, (unsigned)(3 * N));
  deg_kernel<<<blk(E, T), T, 0, stream>>>(srcs[2], dsts[2], wgt[2], outdeg, indeg, E);
  nw_kernel<<<blk(E, T), T, 0, stream>>>(srcs[2], dsts[2], wgt[2], outdeg, indeg, nw, deg, E);
  float* agg3 = bufA;                         // reuse (h_after1 dead)
  fill_zero_kernel<<<blk(32LL * N, T), T, 0, stream>>>(agg3, (unsigned)(32 * N));
  scatter_kernel<32><<<blk((long long)E * 8, T), T, 0, stream>>>(srcs[2], dsts[2], nw, agg2, agg3, E);
  gemm_wmma_kernel<32, 32><<<blk(gemmThreads, T), T, 0, stream>>>(agg3, Wm[2], out, N);
  normalize_kernel<32><<<blk(32LL * N, T), T, 0, stream>>>(out, deg, bm[2], N);
}